// Yolo_DAGNN_25305947308075
// MI455X (gfx1250) — compile-verified
//
#include <hip/hip_runtime.h>

typedef float v2f __attribute__((ext_vector_type(2)));
typedef float v8f __attribute__((ext_vector_type(8)));

#define N_UAV   4096
#define T_STEPS 3
#define FEAT    36
#define N_VIS   6
#define H       256
#define D_IN    18          // T_STEPS * N_VIS
#define D_OUT   2
#define LDS_STRIDE 257      // 256 + 1 pad -> conflict-free column reads (gcd trick)

// ---------------------------------------------------------------------------
// K1: X[j, t*6+o] = vis_feats[j,t,:] @ W_vis[:,o] + b_vis[o]   (4096 x 18)
// ---------------------------------------------------------------------------
__global__ void vis_proj_kernel(const float* __restrict__ vis_feats,
                                const float* __restrict__ W_vis,
                                const float* __restrict__ b_vis,
                                float* __restrict__ X)
{
    int gid = blockIdx.x * blockDim.x + threadIdx.x;
    if (gid >= N_UAV * D_IN) return;
    int j = gid / D_IN;
    int c = gid - j * D_IN;
    int t = c / N_VIS;
    int o = c - t * N_VIS;
    const float* vf = vis_feats + (size_t)(j * T_STEPS + t) * FEAT;
    float s = b_vis[o];
#pragma unroll
    for (int f = 0; f < FEAT; ++f) s += vf[f] * W_vis[f * N_VIS + o];
    X[gid] = s;
}

// ---------------------------------------------------------------------------
// K2: single fused streaming pass over a_nets (201 MB, HBM-bound).
// One block per row i.  Computes msg_t[i,:] and rowsum rs_t[i] for t=0..2 in
// one read of a_nets[i,:,:], then writes
//   x_agg[i] = [ msg2 | msg1*rs2 | msg0*rs1*rs2 ]
// directly into d_out.
// ---------------------------------------------------------------------------
__global__ void agg_kernel(const float* __restrict__ a_nets,
                           const float* __restrict__ X,
                           float* __restrict__ x_agg)
{
    const int i   = blockIdx.x;
    const int tid = threadIdx.x;      // 0..255

    float acc[21];                    // [0..17] = msg_t[o], [18..20] = rowsums
#pragma unroll
    for (int c = 0; c < 21; ++c) acc[c] = 0.f;

    const float* arow = a_nets + (size_t)i * (N_UAV * 3);

    for (int jt = 0; jt < N_UAV / 256; ++jt) {
        const int j = jt * 256 + tid;
        const float a0 = arow[j * 3 + 0];
        const float a1 = arow[j * 3 + 1];
        const float a2 = arow[j * 3 + 2];
        const float* xr = X + j * D_IN;
#pragma unroll
        for (int o = 0; o < N_VIS; ++o) {
            acc[o]      += a0 * xr[o];
            acc[6 + o]  += a1 * xr[6 + o];
            acc[12 + o] += a2 * xr[12 + o];
        }
        acc[18] += a0; acc[19] += a1; acc[20] += a2;
    }

    // wave32 shuffle reduction, then cross-wave via LDS
#pragma unroll
    for (int c = 0; c < 21; ++c) {
#pragma unroll
        for (int s = 16; s > 0; s >>= 1) acc[c] += __shfl_xor(acc[c], s, 32);
    }

    __shared__ float red[8][21];
    __shared__ float tot[21];
    const int wave = tid >> 5, lane = tid & 31;
    if (lane == 0) {
#pragma unroll
        for (int c = 0; c < 21; ++c) red[wave][c] = acc[c];
    }
    __syncthreads();
    if (tid < 21) {
        float s = 0.f;
#pragma unroll
        for (int w = 0; w < 8; ++w) s += red[w][tid];
        tot[tid] = s;
    }
    __syncthreads();
    if (tid < 18) {
        const int o = tid % 6, blk = tid / 6;
        float v;
        if (blk == 0)      v = tot[12 + o];                  // msg2
        else if (blk == 1) v = tot[6 + o] * tot[20];         // msg1 * rs2
        else               v = tot[o] * tot[19] * tot[20];   // msg0 * rs1 * rs2
        x_agg[(size_t)i * D_IN + tid] = v;
    }
}

// ---------------------------------------------------------------------------
// K3: fused MLP, f32 WMMA (V_WMMA_F32_16X16X4_F32).
// Block = 128 threads (4 waves) handles 16 batch rows; activations ping-pong
// in LDS with stride-257 rows. Each wave owns 4 N-tiles of 16 columns.
// ---------------------------------------------------------------------------
__device__ __forceinline__ void dense_wmma_256(const float* actIn,
                                               const float* __restrict__ W,
                                               const float* __restrict__ B,
                                               float* actOut,
                                               int lane, int wave)
{
    const int half = lane >> 4;       // K-half select per ISA A/B layout
    const int l15  = lane & 15;
    for (int nt = wave; nt < 16; nt += 4) {
        const int n = nt * 16 + l15;
        v8f acc = {};
#pragma unroll 8
        for (int kk = 0; kk < 64; ++kk) {
            const int k0 = kk * 4 + half * 2;
            v2f a, b;
            a[0] = actIn[l15 * LDS_STRIDE + k0];
            a[1] = actIn[l15 * LDS_STRIDE + k0 + 1];
            b[0] = W[k0 * H + n];
            b[1] = W[(k0 + 1) * H + n];
            acc = __builtin_amdgcn_wmma_f32_16x16x4_f32(
                false, a, false, b, (short)0, acc, false, false);
        }
        const float bias = B[n];
#pragma unroll
        for (int r = 0; r < 8; ++r) {
            float v = acc[r] + bias;
            actOut[(r + half * 8) * LDS_STRIDE + nt * 16 + l15] = v > 0.f ? v : 0.f;
        }
    }
}

__global__ void mlp_kernel(const float* __restrict__ xagg,
                           const float* __restrict__ W0, const float* __restrict__ B0,
                           const float* __restrict__ W1, const float* __restrict__ B1,
                           const float* __restrict__ W2, const float* __restrict__ B2,
                           const float* __restrict__ Wp, const float* __restrict__ Bp,
                           float* __restrict__ u)
{
    __shared__ float actA[16 * LDS_STRIDE];
    __shared__ float actB[16 * LDS_STRIDE];
    __shared__ float xa[16 * 20];     // 16x18 input tile, K padded to 20

    const int tid  = threadIdx.x;     // 0..127
    const int lane = tid & 31;
    const int wave = tid >> 5;        // 0..3
    const int i0   = blockIdx.x * 16;
    const int half = lane >> 4;
    const int l15  = lane & 15;

    // stage x_agg tile, zero-pad K=18,19
    for (int idx = tid; idx < 16 * 20; idx += 128) {
        const int m = idx / 20, c = idx - m * 20;
        xa[idx] = (c < D_IN) ? xagg[(size_t)(i0 + m) * D_IN + c] : 0.f;
    }
    __syncthreads();

    // ---- layer 0: (16x20) @ W0(18x256), K padded with zeros ----
    for (int nt = wave; nt < 16; nt += 4) {
        const int n = nt * 16 + l15;
        v8f acc = {};
#pragma unroll
        for (int kk = 0; kk < 5; ++kk) {
            const int k0 = kk * 4 + half * 2;
            v2f a, b;
            a[0] = xa[l15 * 20 + k0];
            a[1] = xa[l15 * 20 + k0 + 1];
            b[0] = (k0     < D_IN) ? W0[(k0    ) * H + n] : 0.f;
            b[1] = (k0 + 1 < D_IN) ? W0[(k0 + 1) * H + n] : 0.f;
            acc = __builtin_amdgcn_wmma_f32_16x16x4_f32(
                false, a, false, b, (short)0, acc, false, false);
        }
        const float bias = B0[n];
#pragma unroll
        for (int r = 0; r < 8; ++r) {
            float v = acc[r] + bias;
            actA[(r + half * 8) * LDS_STRIDE + nt * 16 + l15] = v > 0.f ? v : 0.f;
        }
    }
    __syncthreads();

    // ---- layer 1: actA @ W1 -> actB ----
    dense_wmma_256(actA, W1, B1, actB, lane, wave);
    __syncthreads();

    // ---- layer 2: actB @ W2 -> actA ----
    dense_wmma_256(actB, W2, B2, actA, lane, wave);
    __syncthreads();

    // ---- head: (16x256) @ Wp(256x2) + Bp, one output per lane of wave 0 ----
    if (wave == 0) {
        const int m = l15, col = half;
        float s = Bp[col];
#pragma unroll 8
        for (int k = 0; k < H; ++k) s += actA[m * LDS_STRIDE + k] * Wp[k * D_OUT + col];
        u[(size_t)(i0 + m) * D_OUT + col] = s;
    }
}

// ---------------------------------------------------------------------------
extern "C" void kernel_launch(void* const* d_in, const int* in_sizes, int n_in,
                              void* d_out, int out_size, void* d_ws, size_t ws_size,
                              hipStream_t stream)
{
    const float* vis_feats = (const float*)d_in[0];
    const float* a_nets    = (const float*)d_in[1];
    const float* W_vis     = (const float*)d_in[2];
    const float* b_vis     = (const float*)d_in[3];
    const float* W_h0      = (const float*)d_in[4];
    const float* b_h0      = (const float*)d_in[5];
    const float* W_h1      = (const float*)d_in[6];
    const float* b_h1      = (const float*)d_in[7];
    const float* W_h2      = (const float*)d_in[8];
    const float* b_h2      = (const float*)d_in[9];
    const float* W_p       = (const float*)d_in[10];
    const float* b_p       = (const float*)d_in[11];

    float* out       = (float*)d_out;
    float* x_agg_out = out;                       // 4096*18 floats
    float* u_out     = out + (size_t)N_UAV * D_IN; // 4096*2 floats
    float* X         = (float*)d_ws;              // 4096*18 floats scratch

    vis_proj_kernel<<<(N_UAV * D_IN + 255) / 256, 256, 0, stream>>>(
        vis_feats, W_vis, b_vis, X);

    agg_kernel<<<N_UAV, 256, 0, stream>>>(a_nets, X, x_agg_out);

    mlp_kernel<<<N_UAV / 16, 128, 0, stream>>>(
        x_agg_out, W_h0, b_h0, W_h1, b_h1, W_h2, b_h2, W_p, b_p, u_out);
}